// HungarianMatcher_34574486733449
// MI455X (gfx1250) — compile-verified
//
#include <hip/hip_runtime.h>
#include <hip/hip_bf16.h>
#include <math.h>

#define B_ 16
#define N_ 512
#define M_ 256
#define D_ 3
#define MATCH_R_ 0.5f
#define INF_ 1e30f

typedef __attribute__((ext_vector_type(2))) float v2f;
typedef __attribute__((ext_vector_type(8))) float v8f;

// ---------------------------------------------------------------------------
// Kernel 1: distT[b][m][n] = || output[b][n] - targets[b][m] ||
// via V_WMMA_F32_16X16X4_F32 with both norm terms folded into the MMA:
//   A(16x4) rows (targets)   = [-2tx, -2ty, -2tz, 1]
//   B(4x16) cols (outputs)   = [ ox,   oy,   oz, |o|^2]
//   C(16x16) seeded          = |t_row|^2        (broadcast per row via shfl)
//   D = A*B + C = |o|^2 + |t|^2 - 2 o.t  ->  sqrt(relu(.))
// One wave = one 16-target tile swept over 4 output subtiles (4 WMMAs/wave).
// ---------------------------------------------------------------------------
__global__ __launch_bounds__(256)
void hm_dist_wmma_kernel(const float* __restrict__ out_pts,
                         const float* __restrict__ tgt_pts,
                         float* __restrict__ distT) {
  const int wave = (int)((blockIdx.x * blockDim.x + threadIdx.x) >> 5);
  const int lane = (int)(threadIdx.x & 31);
  // 128 waves per batch: 16 m-tiles * 8 n-groups (4 subtiles each)
  const int b   = wave >> 7;
  const int rem = wave & 127;
  const int tm  = rem >> 3;   // m tile 0..15
  const int g   = rem & 7;    // n group 0..7
  const int p   = lane & 15;  // row/col index within tile
  const int hi  = lane >> 4;  // half-wave selects K=0,1 vs K=2,3
  const int m0  = tm * 16;

  // Target point for A-matrix row p (lane p and lane p+16 hold same row)
  const float* t = tgt_pts + ((size_t)b * M_ + (size_t)(m0 + p)) * D_;
  const float tx = t[0], ty = t[1], tz = t[2];
  const float nt2 = tx * tx + ty * ty + tz * tz;

  v2f a;
  if (hi == 0) { a.x = -2.0f * tx; a.y = -2.0f * ty; }  // K=0,1
  else         { a.x = -2.0f * tz; a.y = 1.0f;       }  // K=2,3

  // Seed C with |t_row|^2 once per wave (C-layout row = r + 8*hi); this
  // both hoists the shuffles out of the subtile loop and lets the matrix
  // pipe perform the "+|t|^2" add for free.
  v8f cseed;
#pragma unroll
  for (int r = 0; r < 8; ++r) {
    cseed[r] = __shfl(nt2, r + 8 * hi, 32);
  }

  float* dbase = distT + (size_t)b * M_ * N_;

#pragma unroll
  for (int s = 0; s < 4; ++s) {
    const int n0 = g * 64 + s * 16;
    const float* o = out_pts + ((size_t)b * N_ + (size_t)(n0 + p)) * D_;
    const float ox = o[0], oy = o[1], oz = o[2];
    const float no2 = ox * ox + oy * oy + oz * oz;

    v2f bb;
    if (hi == 0) { bb.x = ox; bb.y = oy;  }  // K=0,1
    else         { bb.x = oz; bb.y = no2; }  // K=2,3

    // (neg_a, A, neg_b, B, c_mod, C, reuse_a, reuse_b)
    v8f c = __builtin_amdgcn_wmma_f32_16x16x4_f32(false, a, false, bb,
                                                  (short)0, cseed, false, false);

#pragma unroll
    for (int r = 0; r < 8; ++r) {
      const int row = r + 8 * hi;                   // target row in tile
      const float d = sqrtf(fmaxf(c[r], 0.0f));     // sqrt(relu(d2))
      dbase[(size_t)(m0 + row) * N_ + (size_t)(n0 + p)] = d;
    }
  }
}

// ---------------------------------------------------------------------------
// Kernel 2: per-batch Jonker-Volgenant LSA on cost = distT[b] (M_ rows x
// N_ cols), one workgroup (512 threads, 16 waves) per batch, state in LDS.
// Then emit (pts, pts_target, labels, match, cnt) concatenated into d_out.
// ---------------------------------------------------------------------------
__global__ __launch_bounds__(512)
void hm_lsa_kernel(const float* __restrict__ out_pts,
                   const float* __restrict__ tgt_pts,
                   const float* __restrict__ distT,
                   float* __restrict__ d_out) {
  __shared__ float s_v[N_ + 1];
  __shared__ float s_minv[N_ + 1];
  __shared__ float s_u[M_ + 1];
  __shared__ int   s_p[N_ + 1];
  __shared__ int   s_way[N_ + 1];
  __shared__ int   s_used[N_ + 1];
  __shared__ float s_redV[16];
  __shared__ int   s_redI[16];
  __shared__ float s_delta;
  __shared__ int   s_j0, s_j1;

  const int tid  = (int)threadIdx.x;
  const int b    = (int)blockIdx.x;
  const int j    = tid + 1;      // column id 1..N_
  const int lane = tid & 31;
  const int wid  = tid >> 5;
  const float* cost = distT + (size_t)b * M_ * N_;

  // ---- init dual variables / assignment ----
  s_p[tid] = 0; s_v[tid] = 0.0f;
  if (tid == 0) { s_p[N_] = 0; s_v[N_] = 0.0f; }
  if (tid <= M_) s_u[tid] = 0.0f;
  __syncthreads();

  for (int i = 1; i <= M_; ++i) {
    s_minv[j] = INF_;
    s_used[j] = 0;
    if (tid == 0) { s_used[0] = 0; s_p[0] = i; s_j0 = 0; }
    __syncthreads();

    for (;;) {
      if (tid == 0) s_used[s_j0] = 1;
      __syncthreads();

      const int   j0  = s_j0;
      const int   i0  = s_p[j0];
      const float ui0 = s_u[i0];
      const float* crow = cost + (size_t)(i0 - 1) * N_;  // coalesced row read

      float val;
      if (!s_used[j]) {
        const float cur = crow[tid] - ui0 - s_v[j];
        if (cur < s_minv[j]) { s_minv[j] = cur; s_way[j] = j0; }
        val = s_minv[j];
      } else {
        val = INF_;
      }
      int idx = j;

      // wave32 argmin butterfly (prefer lowest column index on ties)
#pragma unroll
      for (int off = 16; off > 0; off >>= 1) {
        const float ov = __shfl_down(val, (unsigned)off, 32);
        const int   oi = __shfl_down(idx, (unsigned)off, 32);
        if (ov < val || (ov == val && oi < idx)) { val = ov; idx = oi; }
      }
      if (lane == 0) { s_redV[wid] = val; s_redI[wid] = idx; }
      __syncthreads();
      if (tid == 0) {
        float bv = s_redV[0]; int bi = s_redI[0];
#pragma unroll
        for (int w = 1; w < 16; ++w) {
          const float wv = s_redV[w]; const int wi = s_redI[w];
          if (wv < bv || (wv == bv && wi < bi)) { bv = wv; bi = wi; }
        }
        s_delta = bv; s_j1 = bi;
      }
      __syncthreads();

      const float delta = s_delta;
      const int   j1    = s_j1;
      // dual update: used columns map to distinct rows -> race-free
      if (s_used[j]) { s_u[s_p[j]] += delta; s_v[j] -= delta; }
      else           { s_minv[j] -= delta; }
      if (tid == 0)  { s_u[s_p[0]] += delta; s_v[0] -= delta; s_j0 = j1; }
      __syncthreads();

      if (s_p[j1] == 0) break;
    }

    // augment along alternating path (serial, thread 0)
    if (tid == 0) {
      int j0a = s_j1;
      while (j0a != 0) { const int jw = s_way[j0a]; s_p[j0a] = s_p[jw]; j0a = jw; }
    }
    __syncthreads();
  }

  // ---- emit outputs (one 4-byte slot per reference element, in order) ----
  const int    oi = tid;                       // output index within batch
  const size_t gi = (size_t)b * N_ + (size_t)oi;

  float* pts        = d_out;
  float* pts_target = d_out + (size_t)B_ * N_ * D_;
  float* labels     = d_out + 2 * (size_t)B_ * N_ * D_;
  int*   match      = (int*)(d_out + 2 * (size_t)B_ * N_ * D_ + (size_t)B_ * N_);
  int*   cnt        = (int*)(d_out + 2 * (size_t)B_ * N_ * D_ + 2 * (size_t)B_ * N_);

  pts[gi * 3 + 0] = out_pts[gi * 3 + 0];
  pts[gi * 3 + 1] = out_pts[gi * 3 + 1];
  pts[gi * 3 + 2] = out_pts[gi * 3 + 2];

  const int pi = s_p[j];
  float t0 = 0.0f, t1 = 0.0f, t2 = 0.0f, lab = 0.0f;
  int mi = -1;
  if (pi > 0) {
    const int ti = pi - 1;
    const float dv = cost[(size_t)ti * N_ + (size_t)oi];
    if (dv < MATCH_R_) {
      const float* tp = tgt_pts + ((size_t)b * M_ + (size_t)ti) * D_;
      t0 = tp[0]; t1 = tp[1]; t2 = tp[2];
      lab = 1.0f; mi = ti;
    }
  }
  pts_target[gi * 3 + 0] = t0;
  pts_target[gi * 3 + 1] = t1;
  pts_target[gi * 3 + 2] = t2;
  labels[gi] = lab;
  match[gi]  = mi;
  if (tid == 0) cnt[b] = N_;
}

// ---------------------------------------------------------------------------
extern "C" void kernel_launch(void* const* d_in, const int* in_sizes, int n_in,
                              void* d_out, int out_size, void* d_ws, size_t ws_size,
                              hipStream_t stream) {
  (void)in_sizes; (void)n_in; (void)out_size; (void)ws_size;
  const float* output  = (const float*)d_in[0];   // (B, N, D) f32
  const float* targets = (const float*)d_in[1];   // (B, M, D) f32
  float* distT = (float*)d_ws;                    // (B, M, N) f32, 8 MB

  // 2048 waves total = 16 batches * 16 m-tiles * 8 n-groups; 8 waves/block
  hm_dist_wmma_kernel<<<256, 256, 0, stream>>>(output, targets, distT);
  // one workgroup per batch
  hm_lsa_kernel<<<B_, 512, 0, stream>>>(output, targets, distT, (float*)d_out);
}